// BuildCost_45638322487526
// MI455X (gfx1250) — compile-verified
//
#include <hip/hip_runtime.h>
#include <cstdint>
#include <cstddef>

// ---- problem constants (from reference) ----
#define ANGN   9
#define NCH    4
#define HWDIM  96
#define NDISP  9
#define NCOMBO (NCH * ANGN * ANGN)     // 324 floats per output pixel
#define TPX    32                      // pixels per workgroup tile
#define TILE_F (TPX * NCOMBO)          // 10368 floats per tile
#define WAVE_PIX 4                     // pixels DMA'd per wave
#define WAVE_F (WAVE_PIX * NCOMBO)     // 1296 floats per wave DMA
#define COST_ELEMS 53747712ull         // 2*9*96*96*324
#define CTR_F4 18432                   // 2*4*96*96 / 4

typedef __attribute__((ext_vector_type(4))) unsigned int u32x4;
typedef __attribute__((ext_vector_type(8))) int          i32x8;
typedef __attribute__((ext_vector_type(4))) int          i32x4;

#if __has_include(<hip/amd_detail/amd_gfx1250_TDM.h>)
#define TDM_ARITY6 1
#else
#define TDM_ARITY6 0
#endif

// grid: 2*9*96*3 = 5184 blocks of 256 threads
__global__ __launch_bounds__(256) void build_cost_kernel(
    const float* __restrict__ x, const float* __restrict__ mask,
    float* __restrict__ out)
{
    __shared__ float tile[TPX][NCOMBO];     // 41472 bytes

    const int bx = blockIdx.x;
    const int t  = bx % 3;                  // 32-pixel column tile
    const int r  = (bx / 3) % HWDIM;        // row
    const int d  = (bx / (3 * HWDIM)) % NDISP;
    const int b  = bx / (3 * HWDIM * NDISP);
    const int disp = d - 4;

    const int lane = threadIdx.x & 31;
    const int wave = threadIdx.x >> 5;
    const int cc   = t * TPX + lane;        // pixel column, always < 96

    // ---- phase 1: gather * mask into LDS, record-major per pixel ----
    for (int combo = wave; combo < NCOMBO; combo += 8) {
        const int c = combo / 81;
        const int a = combo - c * 81;       // u*9 + v
        const int u = a / 9;
        const int v = a - u * 9;
        const int row = r  + (4 - u) * disp;
        const int col = cc + (4 - v) * disp;
        float val = 0.0f;
        if ((unsigned)row < (unsigned)HWDIM && (unsigned)col < (unsigned)HWDIM) {
            const float xv = x[(size_t)(((b * NCH + c) * 81 + a)) * (HWDIM * HWDIM)
                               + (size_t)row * HWDIM + col];
            const float mv = mask[(size_t)(b * 81 + a) * (HWDIM * HWDIM)
                                  + (size_t)r * HWDIM + cc];
            val = xv * mv;
        }
        tile[lane][combo] = val;
    }
    __syncthreads();

    // ---- phase 2: each wave DMAs its 4-pixel slice (1296 contiguous floats) ----
    const size_t out_foff =
        ((size_t)((b * NDISP + d) * HWDIM + r) * HWDIM
         + (size_t)(t * TPX + wave * WAVE_PIX)) * NCOMBO;

#if __has_builtin(__builtin_amdgcn_tensor_store_from_lds)
    const unsigned long long gaddr =
        (unsigned long long)(uintptr_t)(out + out_foff);
    // Generic (flat) address of a __shared__ object carries the wave-relative
    // LDS byte offset in its low 32 bits (ISA 10.2).
    // Taking the address also makes the LDS stores above non-dead.
    const unsigned lds_off =
        (unsigned)(uintptr_t)&tile[wave * WAVE_PIX][0];

    // Compiler barrier: the TDM engine reads LDS behind the compiler's back,
    // so pin the LDS stores before the tensor op is issued.
    asm volatile("" : : "s"(lds_off) : "memory");

    // D# group 0: count=1 | lds_addr | global_addr[56:0] | type=2
    u32x4 g0 = { 1u,
                 lds_off,
                 (unsigned)(gaddr & 0xFFFFFFFFull),
                 (unsigned)((gaddr >> 32) & 0x01FFFFFFull) | 0x80000000u };

    // D# group 1: data_size=2 (4B); tensor_dim0=1296; tensor_dim1=1;
    //             tile_dim0=1296; tile_dim1/2 unused; dim0_stride=1296
    i32x8 g1 = { (int)0x00020000,
                 (int)((unsigned)WAVE_F << 16),   // tensor_dim0[15:0] << 16
                 (int)0x00010000,                 // tensor_dim0 hi=0 | tensor_dim1=1
                 (int)((unsigned)WAVE_F << 16),   // tensor_dim1 hi=0 | tile_dim0
                 0,                               // tile_dim1=0, tile_dim2=0
                 (int)WAVE_F,                     // tensor_dim0_stride[31:0]
                 0, 0 };
    i32x4 z4 = { 0, 0, 0, 0 };
#if TDM_ARITY6
    i32x8 z8 = { 0, 0, 0, 0, 0, 0, 0, 0 };
    __builtin_amdgcn_tensor_store_from_lds(g0, g1, z4, z4, z8, 0);
#else
    __builtin_amdgcn_tensor_store_from_lds(g0, g1, z4, z4, 0);
#endif
#if __has_builtin(__builtin_amdgcn_s_wait_tensorcnt)
    __builtin_amdgcn_s_wait_tensorcnt(0);
#endif
#else
    // Fallback: plain coalesced float4 stores from LDS (no TDM builtin available).
    const float4* lsrc = (const float4*)&tile[0][0];
    float4* gdst = (float4*)(out + ((size_t)((b * NDISP + d) * HWDIM + r) * HWDIM
                                    + (size_t)(t * TPX)) * NCOMBO);
    for (int i = threadIdx.x; i < TILE_F / 4; i += 256) gdst[i] = lsrc[i];
    (void)out_foff;
#endif
}

// ctr = x[:, :, 40, :, :]  -> appended after cost in d_out
__global__ __launch_bounds__(256) void ctr_kernel(
    const float* __restrict__ x, float* __restrict__ out)
{
    const int i  = blockIdx.x * 256 + threadIdx.x;   // float4 index, < 18432
    const int bc = i / (HWDIM * HWDIM / 4);          // / 2304
    const int p  = i - bc * (HWDIM * HWDIM / 4);
    const float4* src = (const float4*)x;
    float4* dst = (float4*)(out + COST_ELEMS);
    dst[i] = src[(size_t)(bc * 81 + 40) * (HWDIM * HWDIM / 4) + p];
}

extern "C" void kernel_launch(void* const* d_in, const int* in_sizes, int n_in,
                              void* d_out, int out_size, void* d_ws, size_t ws_size,
                              hipStream_t stream)
{
    const float* x    = (const float*)d_in[0];
    const float* mask = (const float*)d_in[1];
    float* out        = (float*)d_out;

    build_cost_kernel<<<2 * NDISP * HWDIM * 3, 256, 0, stream>>>(x, mask, out);
    ctr_kernel<<<CTR_F4 / 256, 256, 0, stream>>>(x, out);
}